// PaddedToSegments_23691039605161
// MI455X (gfx1250) — compile-verified
//
#include <hip/hip_runtime.h>
#include <hip/hip_bf16.h>
#include <stdint.h>

// Problem shape (fixed by the reference's setup_inputs()).
constexpr int B = 16;
constexpr int T = 2048;   // 2^11
constexpr int D = 1024;   // row = 4 KB of f32
constexpr int CHUNK = 16; // rows per TDM gather descriptor (16-bit index mode)

typedef unsigned int u32x4 __attribute__((ext_vector_type(4)));
typedef unsigned int u32x8 __attribute__((ext_vector_type(8)));

__device__ __forceinline__ unsigned rfl(unsigned x) {
    return (unsigned)__builtin_amdgcn_readfirstlane((int)x);
}

// ---------------------------------------------------------------------------
// Phase 1: per-batch prefix-sum compaction map (wave32 shuffle scans).
// ---------------------------------------------------------------------------
__global__ __launch_bounds__(1024) void scan_compact_kernel(
    const int* __restrict__ mask,
    int* __restrict__ perm,
    int* __restrict__ counts_i,
    float* __restrict__ valid_idx_f,
    float* __restrict__ counts_f)
{
    const int b    = blockIdx.x;
    const int tid  = threadIdx.x;
    const int lane = tid & 31;     // wave32
    const int wave = tid >> 5;     // 32 waves
    const int base = b * T;

    const int t0 = 2 * tid;
    const int t1 = 2 * tid + 1;
    const int m0 = (mask[base + t0] != 0) ? 1 : 0;
    const int m1 = (mask[base + t1] != 0) ? 1 : 0;
    const int c  = m0 + m1;

    valid_idx_f[base + t0] = -1.0f;
    valid_idx_f[base + t1] = -1.0f;

    int incl = c;
    #pragma unroll
    for (int d = 1; d < 32; d <<= 1) {
        int v = __shfl_up(incl, d, 32);
        if (lane >= d) incl += v;
    }

    __shared__ int wsum[32];
    __shared__ int s_total;
    if (lane == 31) wsum[wave] = incl;
    __syncthreads();

    if (wave == 0) {
        int wv    = wsum[lane];
        int wincl = wv;
        #pragma unroll
        for (int d = 1; d < 32; d <<= 1) {
            int v = __shfl_up(wincl, d, 32);
            if (lane >= d) wincl += v;
        }
        wsum[lane] = wincl - wv;
        if (lane == 31) s_total = wincl;
    }
    __syncthreads();

    int pos = wsum[wave] + (incl - c);
    if (m0) {
        perm[base + pos]        = t0;
        valid_idx_f[base + pos] = (float)t0;
        pos++;
    }
    if (m1) {
        perm[base + pos]        = t1;
        valid_idx_f[base + pos] = (float)t1;
    }
    if (tid == 0) {
        counts_i[b] = s_total;
        counts_f[b] = (float)s_total;
    }
}

// ---------------------------------------------------------------------------
// Phase 2: TDM gather. One wave per 16-row output chunk.
//   tensor_load_to_lds (gather mode, 16 x 16-bit strictly-increasing row
//   indices) DMAs up to 16 source rows (64 KB) into LDS in one descriptor;
//   the invalid-tail zero-fill overlaps with the DMA in flight; then
//   s_wait_tensorcnt 0 and tensor_store_from_lds streams the compacted
//   tile out as one 1D tile. Streaming data: NT temporal hints (no reuse,
//   don't pollute L2). s_endpgm's implicit wait-idle drains the store.
// ---------------------------------------------------------------------------
__global__ __launch_bounds__(32) void gather_tdm_kernel(
    const float* __restrict__ in,
    const int* __restrict__ perm,
    const int* __restrict__ counts_i,
    float* __restrict__ out)
{
    __shared__ float buf[CHUNK * D];   // 64 KB staging tile

    const int r    = blockIdx.x;       // chunk id in [0, B*T/CHUNK)
    const int b    = r >> 7;           // / (T/CHUNK = 128)
    const int c    = r & 127;          // chunk within batch
    const int j0   = c * CHUNK;        // first output row of this chunk
    const int lane = threadIdx.x;      // single wave32 per block
    const int base = b * T;

    const int count = counts_i[b];
    int nv = count - j0;
    if (nv < 0) nv = 0;
    if (nv > CHUNK) nv = CHUNK;
    const unsigned nvalid = rfl((unsigned)nv);

    // LDS byte offset of the staging tile (flat LDS addr -> low 32 bits).
    const unsigned lds_off = rfl((unsigned)(unsigned long long)(void*)buf);

    if (nvalid > 0) {
        // ---- pack the (strictly increasing) source row indices, 16-bit each
        unsigned idx[CHUNK];
        unsigned last = 0;
        #pragma unroll
        for (int k = 0; k < CHUNK; ++k) {
            if (k < (int)nvalid) last = (unsigned)perm[base + j0 + k] & 0xFFFFu;
            idx[k] = last;             // beyond tile_dim1: ignored by TDM
        }

        // ---- Load descriptor: gather mode, tensor = inputs[b] (2048x1024 f32)
        const unsigned long long ga =
            (unsigned long long)(in + (size_t)b * T * D);
        u32x4 g0;
        g0[0] = 0x80000001u;                               // gather_mode=1, 16b idx, count=1
        g0[1] = lds_off;                                   // lds_addr
        g0[2] = rfl((unsigned)ga);                         // global_addr[31:0]
        g0[3] = rfl((unsigned)((ga >> 32) & 0x01FFFFFFu))
                | 0x80000000u;                             // addr[56:32] | type=2

        u32x8 g1;
        g1[0] = 2u << 16;                                  // data_size = 4 B
        g1[1] = (unsigned)(D & 0xFFFF) << 16;              // tensor_dim0 lo
        g1[2] = (unsigned)(D >> 16) | ((unsigned)(T & 0xFFFF) << 16); // d0 hi | d1 lo
        g1[3] = (unsigned)(T >> 16) | ((unsigned)D << 16); // d1 hi | tile_dim0 = D
        g1[4] = nvalid;                                    // tile_dim1 = #indices
        g1[5] = (unsigned)D;                               // tensor_dim0_stride lo32
        g1[6] = 0u;                                        // (dim1_stride ignored in gather)
        g1[7] = 0u;

        u32x4 g2, g3;                                      // row indices 0..7 / 8..15
        #pragma unroll
        for (int i = 0; i < 4; ++i) {
            g2[i] = rfl(idx[2 * i]     | (idx[2 * i + 1] << 16));
            g3[i] = rfl(idx[8 + 2 * i] | (idx[9 + 2 * i] << 16));
        }

        // Issue the DMA gather; streaming read -> non-temporal.
        asm volatile("tensor_load_to_lds %0, %1, %2, %3 th:TH_LOAD_NT"
                     :
                     : "s"(g0), "s"(g1), "s"(g2), "s"(g3)
                     : "memory");
    }

    // ---- zero-fill invalid tail rows NOW: overlaps with the TDM gather
    //      in flight (disjoint output rows; STOREcnt-tracked b128 stores).
    {
        const float4 z = make_float4(0.0f, 0.0f, 0.0f, 0.0f);
        for (int k = (int)nvalid; k < CHUNK; ++k) {
            float4* dstrow = (float4*)(out + (size_t)(base + j0 + k) * D);
            #pragma unroll
            for (int i = 0; i < (D / 4) / 32; ++i)         // 8 x b128 per lane
                dstrow[lane + 32 * i] = z;
        }
    }

    if (nvalid > 0) {
        // ---- Store descriptor: 1D tile of nvalid*1024 contiguous elements
        const unsigned nd0 = rfl((unsigned)nvalid * (unsigned)D); // <= 16384
        const unsigned long long oa =
            (unsigned long long)(out + (size_t)(base + j0) * D);
        u32x4 s0;
        s0[0] = 0x00000001u;                               // count=1, no gather
        s0[1] = lds_off;
        s0[2] = rfl((unsigned)oa);
        s0[3] = rfl((unsigned)((oa >> 32) & 0x01FFFFFFu)) | 0x80000000u;

        u32x8 s1;
        s1[0] = 2u << 16;                                  // data_size = 4 B
        s1[1] = (nd0 & 0xFFFFu) << 16;                     // tensor_dim0 lo
        s1[2] = (nd0 >> 16) | (1u << 16);                  // d0 hi | tensor_dim1 = 1
        s1[3] = (nd0 & 0xFFFFu) << 16;                     // d1 hi(0) | tile_dim0 = nd0
        s1[4] = 0u;                                        // tile_dim1 = 0 -> 1D tile
        s1[5] = nd0;                                       // dim0_stride (unused for 1D)
        s1[6] = 0u;
        s1[7] = 0u;

        // Gathered rows must be resident in LDS before the store reads them.
        __builtin_amdgcn_s_wait_tensorcnt(0);

        u32x4 z4 = {0u, 0u, 0u, 0u};                       // tile_dim3/4 = 0 -> unused
        asm volatile("tensor_store_from_lds %0, %1, %2, %3 th:TH_STORE_NT"
                     :
                     : "s"(s0), "s"(s1), "s"(z4), "s"(z4)
                     : "memory");
        // outstanding TENSORcnt drained by implicit wait-idle at s_endpgm
    }
}

extern "C" void kernel_launch(void* const* d_in, const int* in_sizes, int n_in,
                              void* d_out, int out_size, void* d_ws, size_t ws_size,
                              hipStream_t stream) {
    (void)in_sizes; (void)n_in; (void)out_size; (void)ws_size;

    const float* inputs = (const float*)d_in[0];   // [B, T, D] f32
    const int*   mask   = (const int*)d_in[1];     // [B, T] (bool -> int)

    // Outputs concatenated flat in return order, all as f32:
    //   collected [B*T*D] | valid_idx [B*T] | counts [B]
    float* out         = (float*)d_out;
    float* valid_idx_f = out + (size_t)B * T * D;
    float* counts_f    = valid_idx_f + (size_t)B * T;

    // Workspace: int perm table + int counts (~131 KB).
    int* perm     = (int*)d_ws;
    int* counts_i = perm + (size_t)B * T;

    scan_compact_kernel<<<B, 1024, 0, stream>>>(mask, perm, counts_i,
                                                valid_idx_f, counts_f);
    gather_tdm_kernel<<<B * (T / CHUNK), 32, 0, stream>>>(inputs, perm,
                                                          counts_i, out);
}